// GCN_82952998355483
// MI455X (gfx1250) — compile-verified
//
#include <hip/hip_runtime.h>
#include <math.h>

typedef float v2f __attribute__((ext_vector_type(2)));
typedef float v8f __attribute__((ext_vector_type(8)));

// ---------------------------------------------------------------------------
// deg / norm kernels
// ---------------------------------------------------------------------------
__global__ void gcn_deg_init(float* __restrict__ deg, int n) {
  int i = blockIdx.x * blockDim.x + threadIdx.x;
  if (i < n) deg[i] = 1.0f;  // self-loop contribution
}

__global__ void gcn_deg_accum(const long long* __restrict__ dst,
                              float* __restrict__ deg, int ne) {
  int e = blockIdx.x * blockDim.x + threadIdx.x;
  if (e < ne) atomicAdd(&deg[(int)dst[e]], 1.0f);
}

__global__ void gcn_rsqrt_inplace(float* __restrict__ d, int n) {
  int i = blockIdx.x * blockDim.x + threadIdx.x;
  if (i < n) {
    float v = d[i];
    d[i] = v > 0.0f ? rsqrtf(v) : 0.0f;
  }
}

// ---------------------------------------------------------------------------
// WMMA GEMM: Hs = diag(dis) * (A[n x IN] @ W[IN x 64]); also S = Hs
// (S = Hs seeds the scatter accumulator with the self-loop term.)
// One wave -> 16 rows x 64 cols (4 accumulator tiles). W staged in LDS.
// n is a multiple of 16 so EXEC is all-ones inside every computing wave.
// ---------------------------------------------------------------------------
template <int IN>
__global__ __launch_bounds__(128) void gcn_gemm_wmma(
    const float* __restrict__ A, const float* __restrict__ W,
    const float* __restrict__ dis, float* __restrict__ Hs,
    float* __restrict__ S, int rowTiles) {
  constexpr int OUT = 64;
  __shared__ float sW[IN * OUT];
  for (int i = threadIdx.x; i < IN * OUT; i += 128) sW[i] = W[i];
  __syncthreads();

  const int wave = threadIdx.x >> 5;
  const int lane = threadIdx.x & 31;
  const int rowTile = blockIdx.x * 4 + wave;
  if (rowTile >= rowTiles) return;  // wave-uniform; EXEC stays full otherwise

  const int row0 = rowTile * 16;
  const int half2 = (lane >> 4) << 1;  // 0 for lanes 0-15, 2 for lanes 16-31
  const int nCol = lane & 15;
  const float* __restrict__ Arow = A + (size_t)(row0 + nCol) * IN;

  v8f acc0 = {}, acc1 = {}, acc2 = {}, acc3 = {};
  for (int k0 = 0; k0 < IN; k0 += 4) {
    v2f a;
    a.x = Arow[k0 + half2];
    a.y = Arow[k0 + half2 + 1];
    const float* Wk0 = &sW[(k0 + half2) * OUT];
    const float* Wk1 = Wk0 + OUT;
    v2f b0, b1, b2, b3;
    b0.x = Wk0[nCol];      b0.y = Wk1[nCol];
    b1.x = Wk0[16 + nCol]; b1.y = Wk1[16 + nCol];
    b2.x = Wk0[32 + nCol]; b2.y = Wk1[32 + nCol];
    b3.x = Wk0[48 + nCol]; b3.y = Wk1[48 + nCol];
    acc0 = __builtin_amdgcn_wmma_f32_16x16x4_f32(false, a, false, b0, (short)0, acc0, false, false);
    acc1 = __builtin_amdgcn_wmma_f32_16x16x4_f32(false, a, false, b1, (short)0, acc1, false, false);
    acc2 = __builtin_amdgcn_wmma_f32_16x16x4_f32(false, a, false, b2, (short)0, acc2, false, false);
    acc3 = __builtin_amdgcn_wmma_f32_16x16x4_f32(false, a, false, b3, (short)0, acc3, false, false);
  }

  // C/D layout: VGPR v -> row v (lanes 0-15) / row v+8 (lanes 16-31), col = lane&15
  const int rBase = row0 + ((lane >> 4) << 3);
#pragma unroll
  for (int v = 0; v < 8; ++v) {
    const int rr = rBase + v;
    const float s = dis[rr];
    float* __restrict__ hrow = Hs + (size_t)rr * OUT;
    float* __restrict__ srow = S + (size_t)rr * OUT;
    const float x0 = acc0[v] * s, x1 = acc1[v] * s, x2 = acc2[v] * s, x3 = acc3[v] * s;
    hrow[nCol] = x0; hrow[16 + nCol] = x1; hrow[32 + nCol] = x2; hrow[48 + nCol] = x3;
    srow[nCol] = x0; srow[16 + nCol] = x1; srow[32 + nCol] = x2; srow[48 + nCol] = x3;
  }
}

// ---------------------------------------------------------------------------
// Edge scatter-add, 64 features: one thread per (edge, 4-float chunk)
// ---------------------------------------------------------------------------
__global__ void gcn_scatter64(const long long* __restrict__ src,
                              const long long* __restrict__ dst,
                              const float* __restrict__ Hs,
                              float* __restrict__ S, int ne) {
  int t = blockIdx.x * blockDim.x + threadIdx.x;
  if (t >= ne * 16) return;
  const int e = t >> 4;
  const int c = (t & 15) << 2;
  const int sN = (int)src[e];
  const int dN = (int)dst[e];
  const float4 v = *(const float4*)(Hs + (size_t)sN * 64 + c);
  float* sp = S + (size_t)dN * 64 + c;
  atomicAdd(sp + 0, v.x);
  atomicAdd(sp + 1, v.y);
  atomicAdd(sp + 2, v.z);
  atomicAdd(sp + 3, v.w);
}

__global__ void gcn_finalize64(const float* __restrict__ S,
                               const float* __restrict__ dis,
                               const float* __restrict__ b,
                               float* __restrict__ H, int n, int do_tanh) {
  int t = blockIdx.x * blockDim.x + threadIdx.x;
  if (t >= n * 64) return;
  const int i = t >> 6;
  const int f = t & 63;
  float v = S[t] * dis[i] + b[f];
  H[t] = do_tanh ? tanhf(v) : v;
}

// ---------------------------------------------------------------------------
// Layer 2 (64 -> 2) + classifier (2 -> 16): tiny dims, plain VALU
// ---------------------------------------------------------------------------
__global__ void gcn_gemm2(const float* __restrict__ A, const float* __restrict__ W2,
                          const float* __restrict__ dis, float* __restrict__ Hs2,
                          float* __restrict__ S2, int n) {
  int i = blockIdx.x * blockDim.x + threadIdx.x;
  if (i >= n) return;
  const float* a = A + (size_t)i * 64;
  float d0 = 0.f, d1 = 0.f;
#pragma unroll
  for (int k = 0; k < 64; ++k) {
    const float av = a[k];
    d0 = fmaf(av, W2[k * 2 + 0], d0);
    d1 = fmaf(av, W2[k * 2 + 1], d1);
  }
  const float s = dis[i];
  d0 *= s; d1 *= s;
  Hs2[i * 2 + 0] = d0; Hs2[i * 2 + 1] = d1;
  S2[i * 2 + 0] = d0;  S2[i * 2 + 1] = d1;
}

__global__ void gcn_scatter2(const long long* __restrict__ src,
                             const long long* __restrict__ dst,
                             const float* __restrict__ Hs2,
                             float* __restrict__ S2, int ne) {
  int e = blockIdx.x * blockDim.x + threadIdx.x;
  if (e >= ne) return;
  const int sN = (int)src[e];
  const int dN = (int)dst[e];
  const float2 v = *(const float2*)(Hs2 + (size_t)sN * 2);
  atomicAdd(S2 + (size_t)dN * 2 + 0, v.x);
  atomicAdd(S2 + (size_t)dN * 2 + 1, v.y);
}

__global__ void gcn_finalize2(const float* __restrict__ S2,
                              const float* __restrict__ dis,
                              const float* __restrict__ b2,
                              float* __restrict__ emb, int n) {
  int i = blockIdx.x * blockDim.x + threadIdx.x;
  if (i >= n) return;
  const float s = dis[i];
  emb[i * 2 + 0] = tanhf(S2[i * 2 + 0] * s + b2[0]);
  emb[i * 2 + 1] = tanhf(S2[i * 2 + 1] * s + b2[1]);
}

__global__ void gcn_classifier(const float* __restrict__ emb,
                               const float* __restrict__ Wc,
                               const float* __restrict__ bc,
                               float* __restrict__ out, int n) {
  int t = blockIdx.x * blockDim.x + threadIdx.x;
  if (t >= n * 16) return;
  const int i = t >> 4;
  const int c = t & 15;
  const float e0 = emb[i * 2 + 0];
  const float e1 = emb[i * 2 + 1];
  out[t] = fmaf(e0, Wc[c], fmaf(e1, Wc[16 + c], bc[c]));
}

// ---------------------------------------------------------------------------
extern "C" void kernel_launch(void* const* d_in, const int* in_sizes, int n_in,
                              void* d_out, int out_size, void* d_ws, size_t ws_size,
                              hipStream_t stream) {
  const float* x      = (const float*)d_in[0];
  const long long* ei = (const long long*)d_in[1];  // int64 edge_index [2, E]
  const float* W0 = (const float*)d_in[2];
  const float* b0 = (const float*)d_in[3];
  const float* W1 = (const float*)d_in[4];
  const float* b1 = (const float*)d_in[5];
  const float* W2 = (const float*)d_in[6];
  const float* b2 = (const float*)d_in[7];
  const float* Wc = (const float*)d_in[8];
  const float* bc = (const float*)d_in[9];

  const int n  = in_sizes[0] / 128;  // 100000 nodes
  const int ne = in_sizes[1] / 2;    // 1600000 edges
  const long long* src = ei;
  const long long* dst = ei + ne;

  // workspace layout
  float* dis = (float*)d_ws;                    // n floats (deg then rsqrt in-place)
  float* Hs  = dis + 102400;                    // n*64
  float* S   = Hs + (size_t)n * 64;             // n*64
  float* H   = S + (size_t)n * 64;              // n*64
  float* out = (float*)d_out;                   // [n,16]
  float* emb = out + (size_t)n * 16;            // [n,2]

  const int B = 256;
  const int rowTiles = n / 16;                  // 6250, exact
  const dim3 gemmGrid((rowTiles + 3) / 4);

  // normalization
  gcn_deg_init<<<(n + B - 1) / B, B, 0, stream>>>(dis, n);
  gcn_deg_accum<<<(ne + B - 1) / B, B, 0, stream>>>(dst, dis, ne);
  gcn_rsqrt_inplace<<<(n + B - 1) / B, B, 0, stream>>>(dis, n);

  // layer 0: 128 -> 64, no activation
  gcn_gemm_wmma<128><<<gemmGrid, 128, 0, stream>>>(x, W0, dis, Hs, S, rowTiles);
  gcn_scatter64<<<((size_t)ne * 16 + B - 1) / B, B, 0, stream>>>(src, dst, Hs, S, ne);
  gcn_finalize64<<<((size_t)n * 64 + B - 1) / B, B, 0, stream>>>(S, dis, b0, H, n, 0);

  // layer 1: 64 -> 64, tanh
  gcn_gemm_wmma<64><<<gemmGrid, 128, 0, stream>>>(H, W1, dis, Hs, S, rowTiles);
  gcn_scatter64<<<((size_t)ne * 16 + B - 1) / B, B, 0, stream>>>(src, dst, Hs, S, ne);
  gcn_finalize64<<<((size_t)n * 64 + B - 1) / B, B, 0, stream>>>(S, dis, b1, H, n, 1);

  // layer 2: 64 -> 2, tanh -> embeddings (reuse Hs/S low region)
  gcn_gemm2<<<(n + B - 1) / B, B, 0, stream>>>(H, W2, dis, Hs, S, n);
  gcn_scatter2<<<(ne + B - 1) / B, B, 0, stream>>>(src, dst, Hs, S, ne);
  gcn_finalize2<<<(n + B - 1) / B, B, 0, stream>>>(S, dis, b2, emb, n);

  // classifier: 2 -> 16
  gcn_classifier<<<((size_t)n * 16 + B - 1) / B, B, 0, stream>>>(emb, Wc, bc, out, n);
}